// ActorNetwork_61289183314524
// MI455X (gfx1250) — compile-verified
//
#include <hip/hip_runtime.h>
#include <hip/hip_bf16.h>

// ---------------------------------------------------------------------------
// CDNA5 (gfx1250) implementation of the ActorNetwork reference.
// fp32 WMMA (V_WMMA_F32_16X16X4_F32) for all per-row MLPs; f32 atomics for
// the graph scatter-adds. Wave32: one 16-row tile per wave, 8 waves/block.
// ---------------------------------------------------------------------------

typedef __attribute__((ext_vector_type(2))) float v2f;
typedef __attribute__((ext_vector_type(8))) float v8f;

__device__ __forceinline__ v8f wmma4(v2f a, v2f b, v8f c) {
  // D(16x16,f32) = A(16x4,f32) * B(4x16,f32) + C
  return __builtin_amdgcn_wmma_f32_16x16x4_f32(
      /*neg_a=*/false, a, /*neg_b=*/false, b,
      /*c_mod=*/(short)0, c, /*reuse_a=*/false, /*reuse_b=*/false);
}

// ----------------------------- A-matrix functors ---------------------------
// af(row, k) returns element A[row][k] of the (virtual) layer-1 input matrix.

struct AGlobal {            // plain row-major global matrix
  const float* base; int ld;
  __device__ __forceinline__ float operator()(int row, int k) const {
    return base[row * ld + k];
  }
};

struct AOps {               // concat(x_node[32] | y[batch[row]][32] | z[32])
  const float* xn; const float* y; const float* z; const int* batch;
  __device__ __forceinline__ float operator()(int row, int k) const {
    if (k < 32) return xn[row * 32 + k];
    if (k < 64) return y[batch[row] * 32 + (k - 32)];
    return z[k - 64];
  }
};

struct APrlvl {             // concat(limit | y[g][32] | z[32]), K=65 padded to 68
  const float* y; const float* z; int W;
  __device__ __forceinline__ float operator()(int row, int k) const {
    int g = row / W;
    int w = row - g * W;
    if (k == 0)  return (float)(w + 1);
    if (k < 33)  return y[g * 32 + (k - 1)];
    if (k < 65)  return z[k - 33];
    return 0.f;   // K padding 65..67
  }
};

// --------------------------- generic 3-layer MLP ---------------------------
// K0 real input width, KPAD = K0 rounded up to multiple of 4.
// Layer sizes: KPAD -> 32 (relu) -> 16 (relu) -> NOUT (linear).
// One 16-row tile per wave. colsum (optional): atomic per-column sum of out.
template <int K0, int KPAD, int NOUT, typename AF>
__global__ __launch_bounds__(256) void mlp3_kernel(
    AF af, int nrows,
    const float* __restrict__ W1, const float* __restrict__ B1,
    const float* __restrict__ W2, const float* __restrict__ B2,
    const float* __restrict__ W3, const float* __restrict__ B3,
    float* __restrict__ out, float* __restrict__ colsum) {
  constexpr int WAVES = 8;
  constexpr int NT3 = (NOUT + 15) / 16;

  __shared__ float sW1[KPAD * 32];
  __shared__ float sB1[32];
  __shared__ float sW2[32 * 16];
  __shared__ float sB2[16];
  __shared__ float sW3[16 * 32];
  __shared__ float sB3[32];
  __shared__ float sH1[WAVES][16 * 32];
  __shared__ float sH2[WAVES][16 * 16];

  // cooperative weight staging (zero-padded)
  for (int i = threadIdx.x; i < KPAD * 32; i += blockDim.x) {
    int k = i >> 5, n = i & 31;
    sW1[i] = (k < K0) ? W1[k * 32 + n] : 0.f;
  }
  for (int i = threadIdx.x; i < 32 * 16; i += blockDim.x) sW2[i] = W2[i];
  for (int i = threadIdx.x; i < 16 * 32; i += blockDim.x) {
    int k = i >> 5, n = i & 31;
    sW3[i] = (n < NOUT) ? W3[k * NOUT + n] : 0.f;
  }
  if (threadIdx.x < 32) sB1[threadIdx.x] = B1[threadIdx.x];
  if (threadIdx.x < 16) sB2[threadIdx.x] = B2[threadIdx.x];
  if (threadIdx.x < 32)
    sB3[threadIdx.x] = (threadIdx.x < NOUT) ? B3[threadIdx.x] : 0.f;
  __syncthreads();

  const int wave = threadIdx.x >> 5;
  const int lane = threadIdx.x & 31;
  const int m = lane & 15;         // row-in-tile for A / col-in-tile for B,C
  const int hi = lane >> 4;        // half-wave select (K pair / M+8)
  const int rowbase = (blockIdx.x * WAVES + wave) * 16;

  float* h1 = sH1[wave];
  float* h2 = sH2[wave];

  int arow = rowbase + m;
  if (arow >= nrows) arow = nrows - 1;   // clamp; stores are guarded below

  // ---- layer 1: KPAD -> 32, relu -> h1[16][32]
#pragma unroll
  for (int t = 0; t < 2; ++t) {
    v8f c = {};
#pragma unroll 4
    for (int k = 0; k < KPAD; k += 4) {
      const int kk = k + (hi << 1);
      v2f a, b;
      a.x = af(arow, kk);
      a.y = af(arow, kk + 1);
      b.x = sW1[kk * 32 + t * 16 + m];
      b.y = sW1[(kk + 1) * 32 + t * 16 + m];
      c = wmma4(a, b, c);
    }
    const float bm = sB1[t * 16 + m];
#pragma unroll
    for (int r = 0; r < 8; ++r) {
      float v = c[r] + bm;
      v = v > 0.f ? v : 0.f;
      h1[(r + 8 * hi) * 32 + t * 16 + m] = v;
    }
  }
  __syncthreads();

  // ---- layer 2: 32 -> 16, relu -> h2[16][16]
  {
    v8f c = {};
#pragma unroll
    for (int k = 0; k < 32; k += 4) {
      const int kk = k + (hi << 1);
      v2f a, b;
      a.x = h1[m * 32 + kk];
      a.y = h1[m * 32 + kk + 1];
      b.x = sW2[kk * 16 + m];
      b.y = sW2[(kk + 1) * 16 + m];
      c = wmma4(a, b, c);
    }
    const float bm = sB2[m];
#pragma unroll
    for (int r = 0; r < 8; ++r) {
      float v = c[r] + bm;
      v = v > 0.f ? v : 0.f;
      h2[(r + 8 * hi) * 16 + m] = v;
    }
  }
  __syncthreads();

  // ---- layer 3: 16 -> NOUT (linear), guarded global store
#pragma unroll
  for (int t = 0; t < NT3; ++t) {
    v8f c = {};
#pragma unroll
    for (int k = 0; k < 16; k += 4) {
      const int kk = k + (hi << 1);
      v2f a, b;
      a.x = h2[m * 16 + kk];
      a.y = h2[m * 16 + kk + 1];
      b.x = sW3[kk * 32 + t * 16 + m];
      b.y = sW3[(kk + 1) * 32 + t * 16 + m];
      c = wmma4(a, b, c);
    }
    const int col = t * 16 + m;
#pragma unroll
    for (int r = 0; r < 8; ++r) {
      const int row = rowbase + r + 8 * hi;
      if (row < nrows && col < NOUT) {
        const float v = c[r] + sB3[col];
        out[row * NOUT + col] = v;
        if (colsum) atomicAdd(&colsum[col], v);
      }
    }
  }
}

// ------------------------------ small kernels ------------------------------

__global__ void fill_kernel(float* __restrict__ p, float v, int n) {
  int i = blockIdx.x * blockDim.x + threadIdx.x;
  if (i < n) p[i] = v;
}

// deg[col[i]] += 1  (deg pre-initialized to 1.0 to account for self-loops)
__global__ void count_deg_kernel(const int* __restrict__ col, int e,
                                 float* __restrict__ deg) {
  int i = blockIdx.x * blockDim.x + threadIdx.x;
  if (i < e) atomicAdd(&deg[col[i]], 1.0f);
}

__global__ void rsqrt_kernel(float* __restrict__ d, int n) {
  int i = blockIdx.x * blockDim.x + threadIdx.x;
  if (i < n) d[i] = rsqrtf(d[i]);   // deg >= 1 always (self loops)
}

// aggr[i][j] = dinv[i]^2 * h[i][j]   (self-loop term, also initializes aggr)
__global__ void selfloop_kernel(const float* __restrict__ dinv,
                                const float* __restrict__ h,
                                float* __restrict__ aggr, int n) {
  int i = blockIdx.x * blockDim.x + threadIdx.x;
  if (i < n * 8) {
    const int node = i >> 3;
    const float w = dinv[node];
    aggr[i] = w * w * h[i];
  }
}

// aggr[row] += dinv[row]*dinv[col] * h[col]   over E edges (8-wide payload)
__global__ void edge_aggr_kernel(const int* __restrict__ erow,
                                 const int* __restrict__ ecol, int e,
                                 const float* __restrict__ dinv,
                                 const float* __restrict__ h,
                                 float* __restrict__ aggr) {
  int i = blockIdx.x * blockDim.x + threadIdx.x;
  if (i >= e) return;
  const int r = erow[i];
  const int c = ecol[i];
  const float w = dinv[r] * dinv[c];
  const float* hp = h + c * 8;
  float* ap = aggr + r * 8;
#pragma unroll
  for (int j = 0; j < 8; ++j) atomicAdd(&ap[j], w * hp[j]);
}

// ypre[g][0:16] = sum_nodes x ; ypre[g][16:48] = sum_nodes x_node
__global__ __launch_bounds__(256) void ypre_kernel(
    const float* __restrict__ x, const float* __restrict__ xn,
    float* __restrict__ ypre, int opsper) {
  __shared__ float acc[48];
  const int g = blockIdx.x;
  if (threadIdx.x < 48) acc[threadIdx.x] = 0.f;
  __syncthreads();
  const int total = opsper * 48;
  for (int idx = threadIdx.x; idx < total; idx += blockDim.x) {
    const int r = idx / 48;
    const int c = idx - r * 48;
    const int node = g * opsper + r;
    const float v = (c < 16) ? x[node * 16 + c] : xn[node * 32 + (c - 16)];
    atomicAdd(&acc[c], v);   // ds_add_f32
  }
  __syncthreads();
  if (threadIdx.x < 48) ypre[g * 48 + threadIdx.x] = acc[threadIdx.x];
}

// z = mlp_glob(ysum): 32 -> 32 (relu) -> 16 (relu) -> 32. Single tiny row.
__global__ __launch_bounds__(32) void glob_kernel(
    const float* __restrict__ ysum,
    const float* __restrict__ W1, const float* __restrict__ B1,
    const float* __restrict__ W2, const float* __restrict__ B2,
    const float* __restrict__ W3, const float* __restrict__ B3,
    float* __restrict__ z) {
  __shared__ float h1[32];
  __shared__ float h2[16];
  const int t = threadIdx.x;
  {
    float s = B1[t];
    for (int k = 0; k < 32; ++k) s += ysum[k] * W1[k * 32 + t];
    h1[t] = s > 0.f ? s : 0.f;
  }
  __syncthreads();
  if (t < 16) {
    float s = B2[t];
    for (int k = 0; k < 32; ++k) s += h1[k] * W2[k * 16 + t];
    h2[t] = s > 0.f ? s : 0.f;
  }
  __syncthreads();
  {
    float s = B3[t];
    for (int k = 0; k < 16; ++k) s += h2[k] * W3[k * 32 + t];
    z[t] = s;
  }
}

// ------------------------------- launch ------------------------------------

extern "C" void kernel_launch(void* const* d_in, const int* in_sizes, int n_in,
                              void* d_out, int out_size, void* d_ws,
                              size_t ws_size, hipStream_t stream) {
  const float* x     = (const float*)d_in[0];
  const int*   ei    = (const int*)d_in[1];   // [2, E] flattened
  const int*   batch = (const int*)d_in[2];

  const int N = in_sizes[0] / 16;
  const int E = in_sizes[1] / 2;
  const int G = in_sizes[3];
  const int W = (out_size - N) / G;           // 50
  const int OPS = N / G;                      // 200

  // parameter lists, 6 arrays each, in setup_inputs() order
  const float* const* P = (const float* const*)(d_in + 4);
  const float* const* p1 = P + 0;   // mlp1: 16->32->16->8
  const float* const* p2 = P + 6;   // mlp2: 8->32->16->32
  const float* const* pd = P + 12;  // dag : 48->32->16->32
  const float* const* pg = P + 18;  // glob: 32->32->16->32
  const float* const* po = P + 24;  // op  : 96->32->16->1
  const float* const* pp = P + 30;  // prlvl: 65->32->16->1

  // workspace carve-out (floats)
  float* ws    = (float*)d_ws;
  float* dinv  = ws;                       // N      (deg -> dinv in place)
  float* h     = dinv + N;                 // N*8
  float* aggr  = h + (size_t)N * 8;        // N*8
  float* xnode = aggr + (size_t)N * 8;     // N*32
  float* ypre  = xnode + (size_t)N * 32;   // G*48
  float* y     = ypre + (size_t)G * 48;    // G*32
  float* ysum  = y + (size_t)G * 32;       // 32
  float* z     = ysum + 32;                // 32

  float* ops_out   = (float*)d_out;
  float* prlvl_out = ops_out + N;

  const int TB = 256;
  const int tilesN  = (N + 15) / 16;
  const int blocksN = (tilesN + 7) / 8;
  const int tilesG  = (G + 15) / 16;
  const int blocksG = (tilesG + 7) / 8;
  const int rowsP   = G * W;
  const int blocksP = (((rowsP + 15) / 16) + 7) / 8;

  // 1) degree: init deg=1 (self loop), count cols, rsqrt
  fill_kernel<<<(N + TB - 1) / TB, TB, 0, stream>>>(dinv, 1.0f, N);
  fill_kernel<<<1, 64, 0, stream>>>(ysum, 0.0f, 32);
  count_deg_kernel<<<(E + TB - 1) / TB, TB, 0, stream>>>(ei + E, E, dinv);
  rsqrt_kernel<<<(N + TB - 1) / TB, TB, 0, stream>>>(dinv, N);

  // 2) h = mlp1(x)    [N,16] -> [N,8]
  mlp3_kernel<16, 16, 8><<<blocksN, TB, 0, stream>>>(
      AGlobal{x, 16}, N, p1[0], p1[1], p1[2], p1[3], p1[4], p1[5], h, nullptr);

  // 3) GCN aggregation: aggr = sum norm * h[col]
  selfloop_kernel<<<(N * 8 + TB - 1) / TB, TB, 0, stream>>>(dinv, h, aggr, N);
  edge_aggr_kernel<<<(E + TB - 1) / TB, TB, 0, stream>>>(ei, ei + E, E, dinv,
                                                         h, aggr);

  // 4) x_node = mlp2(aggr)   [N,8] -> [N,32]
  mlp3_kernel<8, 8, 32><<<blocksN, TB, 0, stream>>>(
      AGlobal{aggr, 8}, N, p2[0], p2[1], p2[2], p2[3], p2[4], p2[5], xnode,
      nullptr);

  // 5) per-dag sums, dag MLP (fused column-sum for z), global MLP
  ypre_kernel<<<G, TB, 0, stream>>>(x, xnode, ypre, OPS);
  mlp3_kernel<48, 48, 32><<<blocksG, TB, 0, stream>>>(
      AGlobal{ypre, 48}, G, pd[0], pd[1], pd[2], pd[3], pd[4], pd[5], y, ysum);
  glob_kernel<<<1, 32, 0, stream>>>(ysum, pg[0], pg[1], pg[2], pg[3], pg[4],
                                    pg[5], z);

  // 6) op scores: mlp_op(concat(x_node | y[batch] | z)) -> d_out[0:N)
  mlp3_kernel<96, 96, 1><<<blocksN, TB, 0, stream>>>(
      AOps{xnode, y, z, batch}, N, po[0], po[1], po[2], po[3], po[4], po[5],
      ops_out, nullptr);

  // 7) prlvl scores: mlp_prlvl(concat(limit | y | z)) -> d_out[N:N+G*W)
  mlp3_kernel<65, 68, 1><<<blocksP, TB, 0, stream>>>(
      APrlvl{y, z, W}, rowsP, pp[0], pp[1], pp[2], pp[3], pp[4], pp[5],
      prlvl_out, nullptr);
}